// GINetwork_57415122813722
// MI455X (gfx1250) — compile-verified
//
#include <hip/hip_runtime.h>
#include <hip/hip_bf16.h>
#include <math.h>

// Problem constants (from reference setup_inputs)
#define NN 2048   // data rows
#define MM 256    // inducing rows
#define SS 4      // posterior samples
#define FSTR 272  // padded row stride (floats): 17 tiles * 16 >= max D = 257

typedef float v2f __attribute__((ext_vector_type(2)));
typedef float v8f __attribute__((ext_vector_type(8)));

__device__ __forceinline__ float elu1(float x) { return x > 0.f ? x : (expf(x) - 1.f); }

__device__ __forceinline__ unsigned hash32(unsigned x) {
    x ^= x >> 16; x *= 0x7feb352dU;
    x ^= x >> 15; x *= 0x846ca68bU;
    x ^= x >> 16; return x;
}
// Deterministic standard normal from a 32-bit key (Box-Muller)
__device__ __forceinline__ float gauss_hash(unsigned key) {
    unsigned h1 = hash32(key * 2u + 0x9e3779b9u);
    unsigned h2 = hash32(key * 2u + 0x85ebca6bu);
    float u1 = ((float)(h1 >> 8) + 0.5f) * (1.0f / 16777216.0f);
    float u2 = ((float)(h2 >> 8) + 0.5f) * (1.0f / 16777216.0f);
    return sqrtf(-2.f * logf(u1)) * cosf(6.28318530718f * u2);
}

// ---------------------------------------------------------------------------
// Layer-0 input prep: Fn = elu(append_ones(x)) broadcast over S, Un likewise.
// Columns [D, FSTR) are zero-filled so GEMM loads need no bounds checks.
// ---------------------------------------------------------------------------
__global__ void gi_prep_first(const float* __restrict__ x, const float* __restrict__ ip,
                              float* __restrict__ Fn, float* __restrict__ Un) {
    const int D = 9;
    long long idx = (long long)blockIdx.x * blockDim.x + threadIdx.x;
    long long total = (long long)SS * (NN + MM) * FSTR;
    if (idx >= total) return;
    int d = (int)(idx % FSTR);
    long long t = idx / FSTR;
    int r = (int)(t % (NN + MM));
    int s = (int)(t / (NN + MM));
    float v = 0.f;
    if (d < D) {
        float raw;
        if (r < NN) raw = (d < 8) ? x[r * 8 + d] : 1.f;
        else        raw = (d < 8) ? ip[(r - NN) * 8 + d] : 1.f;
        v = elu1(raw); // elu(1) == 1 for the bias column
    }
    if (r < NN) Fn[((long long)s * NN + r) * FSTR + d] = v;
    else        Un[((long long)s * MM + (r - NN)) * FSTR + d] = v;
}

// ---------------------------------------------------------------------------
// Mid-layer prep: Fn = elu(append_ones(Fraw)), Un = elu(append_ones(Uraw)).
// ---------------------------------------------------------------------------
__global__ void gi_prep_mid(const float* __restrict__ Fraw, const float* __restrict__ Uraw,
                            float* __restrict__ Fn, float* __restrict__ Un, int Oprev) {
    const int D = Oprev + 1;
    long long idx = (long long)blockIdx.x * blockDim.x + threadIdx.x;
    long long total = (long long)SS * (NN + MM) * FSTR;
    if (idx >= total) return;
    int d = (int)(idx % FSTR);
    long long t = idx / FSTR;
    int r = (int)(t % (NN + MM));
    int s = (int)(t / (NN + MM));
    float v = 0.f;
    if (r < NN) {
        if (d < D) v = elu1((d < Oprev) ? Fraw[((long long)s * NN + r) * Oprev + d] : 1.f);
        Fn[((long long)s * NN + r) * FSTR + d] = v;
    } else {
        int m = r - NN;
        if (d < D) v = elu1((d < Oprev) ? Uraw[((long long)s * MM + m) * Oprev + d] : 1.f);
        Un[((long long)s * MM + m) * FSTR + d] = v;
    }
}

// ---------------------------------------------------------------------------
// Per-(s,o) posterior solve: one workgroup (16 waves) per matrix.
//   Phase 1: q_prec = Un^T diag(prec) Un + I built in LDS with f32 WMMA.
//            Loads AND stores are unconditional: the LDS matrix is padded to
//            the 16x16 tile grid (272 x 273 f32 ~ 297 KB of the 320 KB WGP
//            LDS -- a CDNA5-only capability).
//   Phase 2: in-LDS Cholesky on the [0,D) block.
//   Phase 3: triangular solves for q_mu and w = q_mu + L^-T eps, KL terms.
// ---------------------------------------------------------------------------
__global__ void gi_solve(const float* __restrict__ Un, const float* __restrict__ logprec,
                         const float* __restrict__ mu, float* __restrict__ W,
                         float* __restrict__ kl_out, int D, int O, unsigned seed) {
    extern __shared__ float lds[];
    const int ROWS = (D + 15) & ~15;           // padded to tile grid
    const int AS = ROWS + 1;                   // LDS row stride
    float* Amat = lds;                         // ROWS x AS
    float* prec = Amat + (size_t)ROWS * AS;    // MM
    float* pm   = prec + MM;                   // MM  (prec * pseud_mu)
    float* v0   = pm + MM;                     // AS  (rhs / eps)
    float* v1   = v0 + AS;                     // AS  (y / z)
    float* v2   = v1 + AS;                     // AS  (q_mu)
    float* red  = v2 + AS;                     // 4 reduction scalars

    const int tid = threadIdx.x, bd = blockDim.x;
    const int s = blockIdx.x / O, o = blockIdx.x % O;
    const float* Ub = Un + (size_t)s * MM * FSTR;

    for (int m = tid; m < MM; m += bd) {
        float p = expf(logprec[(size_t)o * MM + m]);
        prec[m] = p;
        pm[m]   = p * mu[(size_t)o * MM + m];
    }
    if (tid < 4) red[tid] = 0.f;
    __syncthreads();

    // ---- Phase 1: UTLU via v_wmma_f32_16x16x4_f32, tiles striped over waves.
    const int nT = ROWS >> 4;
    const int wave = tid >> 5, lane = tid & 31, row = lane & 15, hi = lane >> 4;
    const int nWaves = bd >> 5;
    for (int t = wave; t < nT * nT; t += nWaves) {
        int d0 = (t / nT) << 4, e0 = (t % nT) << 4;
        int dr = d0 + row, ec = e0 + row;      // A-row index / B-col index
        v8f acc = {};
        for (int mk = 0; mk < MM; mk += 16) {
#pragma unroll
            for (int u = 0; u < 4; ++u) {
                int kb = mk + 4 * u + 2 * hi;
                v2f a, b;
                a.x = Ub[(size_t)kb * FSTR + dr] * prec[kb];
                a.y = Ub[(size_t)(kb + 1) * FSTR + dr] * prec[kb + 1];
                b.x = Ub[(size_t)kb * FSTR + ec];
                b.y = Ub[(size_t)(kb + 1) * FSTR + ec];
                acc = __builtin_amdgcn_wmma_f32_16x16x4_f32(false, a, false, b,
                                                            (short)0, acc, false, false);
            }
        }
        // Unconditional stores: padded LDS rows/cols absorb the tile overhang.
        float* dst = Amat + (size_t)(d0 + 8 * hi) * AS + (e0 + (lane & 15));
#pragma unroll
        for (int g = 0; g < 8; ++g) dst[(size_t)g * AS] = acc[g];
    }
    __syncthreads();
    for (int i = tid; i < D; i += bd) Amat[(size_t)i * AS + i] += 1.0f; // + I/prior_var
    __syncthreads();

    // ---- Phase 2: Cholesky, lower triangle in place
    for (int k = 0; k < D; ++k) {
        if (tid == 0) Amat[(size_t)k * AS + k] = sqrtf(Amat[(size_t)k * AS + k]);
        __syncthreads();
        float dk = Amat[(size_t)k * AS + k];
        for (int i = k + 1 + tid; i < D; i += bd) Amat[(size_t)i * AS + k] /= dk;
        __syncthreads();
        for (int i = k + 1 + tid; i < D; i += bd) {
            float lik = Amat[(size_t)i * AS + k];
            float* Ai = Amat + (size_t)i * AS;
            for (int j = k + 1; j <= i; ++j) Ai[j] -= lik * Amat[(size_t)j * AS + k];
        }
        __syncthreads();
    }

    // ---- Phase 3a: UTLv  -> v0
    for (int d = tid; d < D; d += bd) {
        float acc = 0.f;
        for (int m = 0; m < MM; ++m) acc += Ub[(size_t)m * FSTR + d] * pm[m];
        v0[d] = acc;
    }
    __syncthreads();
    // forward solve L y = v0  -> v1
    for (int j = 0; j < D; ++j) {
        if (tid == 0) v1[j] = v0[j] / Amat[(size_t)j * AS + j];
        __syncthreads();
        float yj = v1[j];
        for (int i = j + 1 + tid; i < D; i += bd) v0[i] -= Amat[(size_t)i * AS + j] * yj;
        __syncthreads();
    }
    // backward solve L^T q_mu = y  -> v2
    for (int j = D - 1; j >= 0; --j) {
        if (tid == 0) v2[j] = v1[j] / Amat[(size_t)j * AS + j];
        __syncthreads();
        float xj = v2[j];
        for (int i = tid; i < j; i += bd) v1[i] -= Amat[(size_t)j * AS + i] * xj;
        __syncthreads();
    }

    // ---- Phase 3b: sample eps, solve L^T z = eps, w = q_mu + z
    for (int d = tid; d < D; d += bd) {
        unsigned key = seed ^ (((unsigned)s * (unsigned)O + (unsigned)o) * 0x01000193u)
                            ^ ((unsigned)d * 0x9e3779b9u);
        v0[d] = gauss_hash(key);
    }
    __syncthreads();
    for (int j = D - 1; j >= 0; --j) {
        if (tid == 0) v1[j] = v0[j] / Amat[(size_t)j * AS + j];
        __syncthreads();
        float xj = v1[j];
        for (int i = tid; i < j; i += bd) v0[i] -= Amat[(size_t)j * AS + i] * xj;
        __syncthreads();
    }
    // w (zero-fill the padded tail so gi_out can load unconditionally)
    for (int d = tid; d < FSTR; d += bd)
        W[((size_t)s * O + o) * FSTR + d] = (d < D) ? (v2[d] + v1[d]) : 0.f;

    // ---- Phase 3c: KL terms
    float logsum = 0.f, mu2 = 0.f, tr = 0.f;
    for (int i = tid; i < D; i += bd) {
        logsum += logf(Amat[(size_t)i * AS + i]);
        mu2 += v2[i] * v2[i];
    }
    // tr((L L^T)^-1) = ||L^-1||_F^2 : per-thread column solves of L x = e_t
    {
        float z[272];
        for (int t = tid; t < D; t += bd) {
            for (int i = t; i < D; ++i) {
                float acc = (i == t) ? 1.f : 0.f;
                const float* Ai = Amat + (size_t)i * AS;
                for (int j = t; j < i; ++j) acc -= Ai[j] * z[j];
                float zi = acc / Ai[i];
                z[i] = zi;
                tr += zi * zi;
            }
        }
    }
    atomicAdd(&red[0], logsum);
    atomicAdd(&red[1], mu2);
    atomicAdd(&red[2], tr);
    __syncthreads();
    if (tid == 0) {
        // prior_var = 1: kl = 0.5*(tr + |mu|^2 - D + 2*sum(log diag L))
        float kl = 0.5f * (red[2] + red[1] - (float)D + 2.f * red[0]);
        atomicAdd(&kl_out[s], kl);
    }
}

// ---------------------------------------------------------------------------
// Output GEMM: Fraw[s,n,o] = Fn[s,n,:].W[s,o,:], Uraw likewise.
// One wave per 16x16 tile; K-pairs are contiguous in memory for BOTH operands
// so each fetch is a single 8-byte (global_load_b64) load.
// ---------------------------------------------------------------------------
__global__ void gi_out(const float* __restrict__ Fn, const float* __restrict__ Un,
                       const float* __restrict__ W, float* __restrict__ Fdst,
                       float* __restrict__ Udst, int D, int O, int finalF) {
    const int s = blockIdx.z;
    const int r0 = blockIdx.x << 4;   // over NN+MM rows (both multiples of 16)
    const int o0 = blockIdx.y << 4;
    const int lane = threadIdx.x & 31, row = lane & 15, hi = lane >> 4;
    const bool isU = (r0 >= NN);
    const float* src = isU ? (Un + (size_t)s * MM * FSTR + (size_t)(r0 - NN) * FSTR)
                           : (Fn + (size_t)s * NN * FSTR + (size_t)r0 * FSTR);
    const float* Wb = W + (size_t)s * 256 * FSTR;  // 256 W rows per sample
    v8f acc = {};
    const int ec = o0 + row;           // W rows [O,16) zeroed for the O==1 layer
    const int DP = (D + 15) & ~15;     // padded K extent (multiple of 16)
    const float* arow = src + (size_t)row * FSTR;
    const float* brow = Wb + (size_t)ec * FSTR;
    for (int kc = 0; kc < DP; kc += 16) {
#pragma unroll
        for (int u = 0; u < 4; ++u) {
            int kb = kc + 4 * u + 2 * hi;
            v2f a = *(const v2f*)(arow + kb);   // 8B-aligned: kb even, FSTR even
            v2f b = *(const v2f*)(brow + kb);
            acc = __builtin_amdgcn_wmma_f32_16x16x4_f32(false, a, false, b,
                                                        (short)0, acc, false, false);
        }
    }
    const int oc = o0 + (lane & 15);
    if (!finalF) {
        // O is a multiple of 16 here: unconditional, coalesced b32 stores.
        if (isU) {
            float* dst = Udst + ((size_t)s * MM + (r0 - NN) + 8 * hi) * O + oc;
#pragma unroll
            for (int g = 0; g < 8; ++g) dst[(size_t)g * O] = acc[g];
        } else {
            float* dst = Fdst + ((size_t)s * NN + r0 + 8 * hi) * O + oc;
#pragma unroll
            for (int g = 0; g < 8; ++g) dst[(size_t)g * O] = acc[g];
        }
    } else if (!isU && oc == 0) {       // final layer: O == 1, F rows only
#pragma unroll
        for (int g = 0; g < 8; ++g)
            Fdst[(size_t)s * NN + r0 + 8 * hi + g] = acc[g];
    }
}

// ---------------------------------------------------------------------------
extern "C" void kernel_launch(void* const* d_in, const int* in_sizes, int n_in,
                              void* d_out, int out_size, void* d_ws, size_t ws_size,
                              hipStream_t stream) {
    const float* x   = (const float*)d_in[0];
    const float* ip  = (const float*)d_in[1];
    const float* mu0 = (const float*)d_in[2];
    const float* lp0 = (const float*)d_in[3];
    const float* mu1 = (const float*)d_in[4];
    const float* lp1 = (const float*)d_in[5];
    const float* mu2 = (const float*)d_in[6];
    const float* lp2 = (const float*)d_in[7];
    float* out = (float*)d_out;

    // Workspace layout (~21 MB of f32, all L2-resident on MI455X's 192 MB L2)
    float* Fn   = (float*)d_ws;
    float* Un   = Fn   + (size_t)SS * NN * FSTR;
    float* Fraw = Un   + (size_t)SS * MM * FSTR;
    float* Uraw = Fraw + (size_t)SS * NN * 256;
    float* W    = Uraw + (size_t)SS * MM * 256;   // SS * 256 * FSTR floats
    float* kl   = out + (size_t)SS * NN;          // KL lives at tail of d_out

    hipMemsetAsync(kl, 0, SS * sizeof(float), stream);

    auto shmem = [](int D) {
        int rows = (D + 15) & ~15, as = rows + 1;
        return (size_t)(rows * as + 2 * MM + 3 * as + 8) * sizeof(float);
    };
    // Allow > 64 KB dynamic LDS (CDNA5 WGP has 320 KB)
    (void)hipFuncSetAttribute(reinterpret_cast<const void*>(&gi_solve),
                              hipFuncAttributeMaxDynamicSharedMemorySize,
                              (int)shmem(257));

    const long long prepTot = (long long)SS * (NN + MM) * FSTR;
    const unsigned prepBlocks = (unsigned)((prepTot + 255) / 256);

    // ---- Layer 0: D = 9, O = 256
    gi_prep_first<<<prepBlocks, 256, 0, stream>>>(x, ip, Fn, Un);
    gi_solve<<<SS * 256, 512, shmem(9), stream>>>(Un, lp0, mu0, W, kl, 9, 256, 1u);
    {
        dim3 g((NN + MM) / 16, 256 / 16, SS);
        gi_out<<<g, 32, 0, stream>>>(Fn, Un, W, Fraw, Uraw, 9, 256, 0);
    }
    // ---- Layer 1: D = 257, O = 256
    gi_prep_mid<<<prepBlocks, 256, 0, stream>>>(Fraw, Uraw, Fn, Un, 256);
    gi_solve<<<SS * 256, 512, shmem(257), stream>>>(Un, lp1, mu1, W, kl, 257, 256, 2u);
    {
        dim3 g((NN + MM) / 16, 256 / 16, SS);
        gi_out<<<g, 32, 0, stream>>>(Fn, Un, W, Fraw, Uraw, 257, 256, 0);
    }
    // ---- Layer 2: D = 257, O = 1, writes final F to d_out
    gi_prep_mid<<<prepBlocks, 256, 0, stream>>>(Fraw, Uraw, Fn, Un, 256);
    // zero W so phantom output rows 1..15 read as 0 in gi_out's B tiles
    hipMemsetAsync(W, 0, (size_t)SS * 256 * FSTR * sizeof(float), stream);
    gi_solve<<<SS * 1, 512, shmem(257), stream>>>(Un, lp2, mu2, W, kl, 257, 1, 3u);
    {
        dim3 g((NN + MM) / 16, 1, SS);
        gi_out<<<g, 32, 0, stream>>>(Fn, Un, W, out, nullptr, 257, 1, 1);
    }
}